// GATLayerSpatial_61435212202105
// MI455X (gfx1250) — compile-verified
//
#include <hip/hip_runtime.h>

// GAT-spatial fused kernel for MI455X (gfx1250, wave32, WGP 320KB LDS).
// One workgroup per (n, h) row: h read once from HBM, out written once.
// All three 32x32x32-class GEMMs run on V_WMMA_F32_16X16X4_F32.

#define ALPHA 0.2f

typedef float v2f __attribute__((ext_vector_type(2)));
typedef float v8f __attribute__((ext_vector_type(8)));

__device__ __forceinline__ v8f wmma4(v2f a, v2f b, v8f c) {
  // 8-arg pattern: (neg_a, A, neg_b, B, c_mod, C, reuse_a, reuse_b)
  return __builtin_amdgcn_wmma_f32_16x16x4_f32(false, a, false, b, (short)0, c,
                                               false, false);
}

// LDS layout in floats
#define OFF_WH   0        // Wh  [w=64][j=32][f=32]          256 KB
#define OFF_E1   65536    // e1  [w=64][i=32]                  8 KB
#define OFF_E2   67584    // e2  [w=64][j=32]                  8 KB
#define OFF_M    69632    // max [j=32][i=32]                  4 KB
#define OFF_S    70656    // 1/sum [j=32][i=32]                4 KB
#define OFF_ADJ  71680    // adj_n [i=32][v=32]                4 KB
#define OFF_SCR  72704    // per-wave hp scratch [8][32][32]  32 KB (+phase-0 temps)
#define SMEM_FLOATS 80896 // 323584 bytes <= 320 KB WGP LDS

__global__ __launch_bounds__(256)
void gat_fused_kernel(const float* __restrict__ hglob,   // (8,64,64,32,32)
                      const float* __restrict__ Wmap,    // (32,32)
                      const float* __restrict__ bmap,    // (32,)
                      const float* __restrict__ a_attn,  // (64,1)
                      const float* __restrict__ Badj,    // (32,32)
                      float* __restrict__ out) {         // (8,64,64,32,32)
  extern __shared__ float smem[];
  const int tid  = threadIdx.x;
  const int lane = tid & 31;
  const int wave = tid >> 5;
  const int r    = lane & 15;       // N / M-within-half index
  const int hihalf = lane >> 4;     // 0: lanes 0-15, 1: lanes 16-31
  const int koff = hihalf * 2;      // K offset for A/B fragments
  const int moff = hihalf * 8;      // row offset for C/D fragments

  const int nb   = blockIdx.x >> 6; // n
  const int hrow = blockIdx.x & 63; // h

  float* Wh   = smem + OFF_WH;
  float* e1   = smem + OFF_E1;
  float* e2   = smem + OFF_E2;
  float* mjs  = smem + OFF_M;
  float* sjs  = smem + OFF_S;
  float* adjl = smem + OFF_ADJ;

  // ---------------- Phase 0: adj_n = D^-1/2 (minmax(B+I)) D^-1/2 ----------
  {
    float* red  = smem + OFF_SCR;        // 256
    float* red2 = red + 256;             // 256
    float* d12  = red + 512;             // 32
    float* al   = red + 544;             // 64: a1|a2 (persists until phase 3)

    float lmin = 1e30f, lmax = -1e30f;
#pragma unroll
    for (int c = 0; c < 4; ++c) {
      int p = tid + 256 * c;                       // p = i*32 + v
      float val = Badj[p] + (((p >> 5) == (p & 31)) ? 1.0f : 0.0f);
      adjl[p] = val;
      lmin = fminf(lmin, val);
      lmax = fmaxf(lmax, val);
    }
    red[tid] = lmin; red2[tid] = lmax;
    __syncthreads();
    for (int s = 128; s > 0; s >>= 1) {
      if (tid < s) {
        red[tid]  = fminf(red[tid],  red[tid + s]);
        red2[tid] = fmaxf(red2[tid], red2[tid + s]);
      }
      __syncthreads();
    }
    const float amin = red[0];
    const float sc   = 1.0f / (red2[0] - amin);
#pragma unroll
    for (int c = 0; c < 4; ++c) {
      int p = tid + 256 * c;
      adjl[p] = (adjl[p] - amin) * sc;
    }
    __syncthreads();
    if (tid < 32) {
      float rs = 0.0f;
      for (int v = 0; v < 32; ++v) rs += adjl[tid * 32 + v];
      d12[tid] = 1.0f / sqrtf(rs);
    }
    if (tid < 64) al[tid] = a_attn[tid];   // a1[0:32] | a2[0:32]
    __syncthreads();
#pragma unroll
    for (int c = 0; c < 4; ++c) {
      int p = tid + 256 * c;
      adjl[p] *= d12[p >> 5] * d12[p & 31];
    }
    __syncthreads();
  }

  // ---------------- Phase 1: Wh = hv @ W_map + b (WMMA f32) ---------------
  // Per wave: 8 w-values; per w: D(32j x 32f) = A(32j x 32t) * B(32t x 32f)
  const float* hblk = hglob + ((size_t)nb * 64 + hrow) * 65536; // [w][t][v]
  {
    v2f bW[8][2];
#pragma unroll
    for (int ko = 0; ko < 8; ++ko) {
      int t = ko * 4 + koff;
#pragma unroll
      for (int nh = 0; nh < 2; ++nh) {
        bW[ko][nh].x = Wmap[t * 32 + nh * 16 + r];
        bW[ko][nh].y = Wmap[(t + 1) * 32 + nh * 16 + r];
      }
    }
    float bm0 = bmap[r], bm1 = bmap[16 + r];

    for (int q = 0; q < 8; ++q) {
      int w = wave * 8 + q;
      const float* hw = hblk + w * 1024;  // [t][v]
      if (q < 7) __builtin_prefetch(hblk + (w + 1) * 1024 + lane * 32, 0, 1);
#pragma unroll
      for (int mh = 0; mh < 2; ++mh) {
        int j = mh * 16 + r;
        v2f afr[8];
#pragma unroll
        for (int ko = 0; ko < 8; ++ko) {
          int t = ko * 4 + koff;
          afr[ko].x = hw[t * 32 + j];          // A[j, t] = h[..,t,j]
          afr[ko].y = hw[(t + 1) * 32 + j];
        }
#pragma unroll
        for (int nh = 0; nh < 2; ++nh) {
          float bmv = nh ? bm1 : bm0;
          v8f c;
#pragma unroll
          for (int g = 0; g < 8; ++g) c[g] = bmv;   // + b_map[f]
#pragma unroll
          for (int ko = 0; ko < 8; ++ko) c = wmma4(afr[ko], bW[ko][nh], c);
#pragma unroll
          for (int g = 0; g < 8; ++g)
            Wh[(w * 32 + mh * 16 + moff + g) * 32 + nh * 16 + r] = c[g];
        }
      }
    }
  }
  __syncthreads();

  // ---------------- Phase 2a: e1/e2 projections ---------------------------
  {
    const float* al = smem + OFF_SCR + 544;
    for (int q = 0; q < 8; ++q) {
      int w = wave * 8 + q;
      float s1 = 0.0f, s2 = 0.0f;
      for (int f = 0; f < 32; ++f) {
        float x = Wh[(w * 32 + lane) * 32 + f];
        s1 += x * al[f];
        s2 += x * al[32 + f];
      }
      e1[w * 32 + lane] = s1;
      e2[w * 32 + lane] = s2;
    }
  }
  __syncthreads();

  // ---------------- Phase 2b: softmax-over-w stats per (i,j) --------------
#pragma unroll
  for (int c = 0; c < 4; ++c) {
    int p = tid + 256 * c;       // p = j*32 + i
    int i = p & 31, j = p >> 5;
    float mmax = -3.4e38f;
    for (int w = 0; w < 64; ++w) {
      float x = e1[w * 32 + i] + e2[w * 32 + j];
      x = x > 0.0f ? x : ALPHA * x;
      mmax = fmaxf(mmax, x);
    }
    float ssum = 0.0f;
    for (int w = 0; w < 64; ++w) {
      float x = e1[w * 32 + i] + e2[w * 32 + j];
      x = x > 0.0f ? x : ALPHA * x;
      ssum += expf(x - mmax);
    }
    mjs[p] = mmax;
    sjs[p] = 1.0f / ssum;
  }
  __syncthreads();

  // ---------------- Phase 3: hp = attn@Wh ; out = hp^T @ adj_n ; ELU ------
  float* scr = smem + OFF_SCR + wave * 1024;   // per-wave hp [i=32][f=32]
  v2f b3[8][2];                                 // adj_n B-fragments
#pragma unroll
  for (int ko = 0; ko < 8; ++ko) {
    int ik = ko * 4 + koff;
#pragma unroll
    for (int nh = 0; nh < 2; ++nh) {
      b3[ko][nh].x = adjl[ik * 32 + nh * 16 + r];
      b3[ko][nh].y = adjl[(ik + 1) * 32 + nh * 16 + r];
    }
  }
  const int hh = hrow >> 5, fl = hrow & 31;

  for (int q = 0; q < 8; ++q) {
    int w = wave * 8 + q;
    v2f b2[8][2];
    float e2v[8][2];
#pragma unroll
    for (int ko = 0; ko < 8; ++ko) {
      int jk = ko * 4 + koff;
#pragma unroll
      for (int nh = 0; nh < 2; ++nh) {
        b2[ko][nh].x = Wh[(w * 32 + jk) * 32 + nh * 16 + r];
        b2[ko][nh].y = Wh[(w * 32 + jk + 1) * 32 + nh * 16 + r];
      }
      e2v[ko][0] = e2[w * 32 + jk];
      e2v[ko][1] = e2[w * 32 + jk + 1];
    }
    // hp[i][f] = sum_j attn[i][j] * Wh[w][j][f]
#pragma unroll
    for (int mh = 0; mh < 2; ++mh) {
      int irow = mh * 16 + r;
      float e1v = e1[w * 32 + irow];
      v2f afr[8];
#pragma unroll
      for (int ko = 0; ko < 8; ++ko) {
        int jk = ko * 4 + koff;
        float x0 = e1v + e2v[ko][0]; x0 = x0 > 0.0f ? x0 : ALPHA * x0;
        float x1 = e1v + e2v[ko][1]; x1 = x1 > 0.0f ? x1 : ALPHA * x1;
        afr[ko].x = expf(x0 - mjs[jk * 32 + irow]) * sjs[jk * 32 + irow];
        afr[ko].y = expf(x1 - mjs[(jk + 1) * 32 + irow]) * sjs[(jk + 1) * 32 + irow];
      }
#pragma unroll
      for (int nh = 0; nh < 2; ++nh) {
        v8f c;
#pragma unroll
        for (int g = 0; g < 8; ++g) c[g] = 0.0f;
#pragma unroll
        for (int ko = 0; ko < 8; ++ko) c = wmma4(afr[ko], b2[ko][nh], c);
#pragma unroll
        for (int g = 0; g < 8; ++g)
          scr[(mh * 16 + moff + g) * 32 + nh * 16 + r] = c[g];
      }
    }
    // out[f][v] = sum_i hp[i][f] * adj_n[i][v]; write with index scramble:
    // out[n, h'=w, w'=2f+hh, f'=fl, v], then ELU.
    float* outw = out + (((size_t)nb * 64 + w) * 65536) + (size_t)fl * 32;
#pragma unroll
    for (int mh = 0; mh < 2; ++mh) {            // f-half
      v2f afr[8];
#pragma unroll
      for (int ko = 0; ko < 8; ++ko) {
        int ik = ko * 4 + koff;
        afr[ko].x = scr[ik * 32 + mh * 16 + r];       // A[f, i] = hp[i][f]
        afr[ko].y = scr[(ik + 1) * 32 + mh * 16 + r];
      }
#pragma unroll
      for (int nh = 0; nh < 2; ++nh) {          // v-half
        v8f c;
#pragma unroll
        for (int g = 0; g < 8; ++g) c[g] = 0.0f;
#pragma unroll
        for (int ko = 0; ko < 8; ++ko) c = wmma4(afr[ko], b3[ko][nh], c);
#pragma unroll
        for (int g = 0; g < 8; ++g) {
          int frow = mh * 16 + moff + g;
          float v = c[g];
          v = v > 0.0f ? v : (expf(v) - 1.0f);  // ELU(alpha=1)
          outw[(size_t)(2 * frow + hh) * 1024 + nh * 16 + r] = v;
        }
      }
    }
  }
}

extern "C" void kernel_launch(void* const* d_in, const int* in_sizes, int n_in,
                              void* d_out, int out_size, void* d_ws, size_t ws_size,
                              hipStream_t stream) {
  (void)in_sizes; (void)n_in; (void)out_size; (void)d_ws; (void)ws_size;
  const float* h      = (const float*)d_in[0];
  const float* W_map  = (const float*)d_in[1];
  const float* b_map  = (const float*)d_in[2];
  const float* a_attn = (const float*)d_in[3];
  const float* B_adj  = (const float*)d_in[4];
  float* out = (float*)d_out;

  const size_t shmem = (size_t)SMEM_FLOATS * sizeof(float); // 323584 B
  gat_fused_kernel<<<dim3(8 * 64), dim3(256), shmem, stream>>>(
      h, W_map, b_map, a_attn, B_adj, out);
}